// DeeperRGCN_84413287236349
// MI455X (gfx1250) — compile-verified
//
#include <hip/hip_runtime.h>
#include <hip/hip_bf16.h>

typedef __attribute__((ext_vector_type(2))) float v2f;
typedef __attribute__((ext_vector_type(8))) float v8f;

__device__ __forceinline__ unsigned short f2bf(float f) {
    unsigned u = __float_as_uint(f);
    unsigned r = u + 0x7FFFu + ((u >> 16) & 1u);   // round-to-nearest-even
    return (unsigned short)(r >> 16);
}
__device__ __forceinline__ float bf2f(unsigned short h) {
    return __uint_as_float(((unsigned)h) << 16);
}

// ---------------------------------------------------------------------------
// Setup kernels: degree per (dst,rel), in-degree count, CSR build
// ---------------------------------------------------------------------------
__global__ void deg_cnt_kernel(const int* __restrict__ ei, const int* __restrict__ et,
                               float* __restrict__ deg, int* __restrict__ cnt,
                               int E, int R) {
    int e = blockIdx.x * blockDim.x + threadIdx.x;
    if (e < E) {
        int dst = ei[E + e];
        int r   = et[e];
        atomicAdd(&deg[dst * R + r], 1.0f);
        atomicAdd(&cnt[dst], 1);
    }
}

// single-block prefix scan over cnt -> offsets, cursor
__global__ void scan_kernel(const int* __restrict__ cnt, int* __restrict__ offsets,
                            int* __restrict__ cursor, int N) {
    __shared__ int sdata[1024];
    __shared__ int carry;
    int t = threadIdx.x;
    if (t == 0) { carry = 0; offsets[0] = 0; }
    __syncthreads();
    for (int base = 0; base < N; base += 1024) {
        int i = base + t;
        int v = (i < N) ? cnt[i] : 0;
        sdata[t] = v;
        __syncthreads();
        for (int off = 1; off < 1024; off <<= 1) {
            int x = (t >= off) ? sdata[t - off] : 0;
            __syncthreads();
            sdata[t] += x;
            __syncthreads();
        }
        int inc = sdata[t] + carry;
        if (i < N) { offsets[i + 1] = inc; cursor[i] = inc - v; }
        __syncthreads();
        if (t == 1023) carry = inc;
        __syncthreads();
    }
}

__global__ void scatter_kernel(const int* __restrict__ ei, const int* __restrict__ et,
                               const float* __restrict__ deg, int* __restrict__ cursor,
                               int* __restrict__ ssrc, int* __restrict__ srel,
                               float* __restrict__ snorm, int E, int R) {
    int e = blockIdx.x * blockDim.x + threadIdx.x;
    if (e < E) {
        int src = ei[e];
        int dst = ei[E + e];
        int r   = et[e];
        int pos = atomicAdd(&cursor[dst], 1);
        ssrc[pos]  = src;
        srel[pos]  = r;
        snorm[pos] = 1.0f / deg[dst * R + r];
    }
}

// ---------------------------------------------------------------------------
// WMMA helpers: f32 16x16x4, A frag from LDS rows, B frag from global weight
// ---------------------------------------------------------------------------
__device__ __forceinline__ v8f wmma4(float aX, float aY, v2f b, v8f c) {
    v2f a;
    a.x = aX;
    a.y = aY;
    return __builtin_amdgcn_wmma_f32_16x16x4_f32(false, a, false, b, (short)0, c,
                                                 false, false);
}

// Accumulate 4 row-tiles (64 rows) x one 16-col tile against B (128x128 row-major).
__device__ __forceinline__ void gemm_tile4(const float (*ldsA)[132],
                                           const float* __restrict__ B,
                                           int m, int half, int col0,
                                           v8f& c0, v8f& c1, v8f& c2, v8f& c3) {
    for (int k0 = 0; k0 < 128; k0 += 4) {
        int kk = k0 + 2 * half;
        const float* bp = B + (size_t)kk * 128 + col0 + m;
        v2f b;
        b.x = bp[0];
        b.y = bp[128];
        c0 = wmma4(ldsA[m     ][kk], ldsA[m     ][kk + 1], b, c0);
        c1 = wmma4(ldsA[m + 16][kk], ldsA[m + 16][kk + 1], b, c1);
        c2 = wmma4(ldsA[m + 32][kk], ldsA[m + 32][kk + 1], b, c2);
        c3 = wmma4(ldsA[m + 48][kk], ldsA[m + 48][kk + 1], b, c3);
    }
}

// ---------------------------------------------------------------------------
// Fused per-layer GEMM: block stages 64 rows of P once in LDS, then computes
//   Q[n,:]           = P[n,:] @ root + bias            (f32)
//   XW[n, rel*128+:] = P[n,:] @ W_rel                  (bf16), rel = 0..R-1
// block = 256 threads = 8 waves; wave w owns 16-col tile w across all outputs.
// ---------------------------------------------------------------------------
__global__ void gemm_layer_kernel(const float* __restrict__ P,
                                  const float* __restrict__ Wrel,   // [R,128,128]
                                  const float* __restrict__ root,   // [128,128]
                                  const float* __restrict__ bias,   // [128]
                                  float* __restrict__ Q,            // [N,128]
                                  unsigned short* __restrict__ XW,  // [N, R*128]
                                  int N, int R) {
    __shared__ float ldsA[64][132];
    int  row0 = blockIdx.x * 64;
    int  t    = threadIdx.x;
    bool full = (row0 + 64 <= N);   // uniform across block

    if (full) {
        for (int idx = t; idx < 64 * 128; idx += 256) {
            int r = idx >> 7, c = idx & 127;
            ldsA[r][c] = P[(size_t)(row0 + r) * 128 + c];
        }
    } else {
        for (int idx = t; idx < 64 * 128; idx += 256) {
            int r = idx >> 7, c = idx & 127;
            int gr = row0 + r;
            ldsA[r][c] = (gr < N) ? P[(size_t)gr * 128 + c] : 0.0f;
        }
    }
    __syncthreads();

    int wave = t >> 5;          // 16-col tile 0..7
    int lane = t & 31;
    int half = lane >> 4;
    int m    = lane & 15;
    int col0 = wave * 16;
    int ldx  = R * 128;

    // ---- root part: Q = P @ root + bias ----
    {
        float bv = bias[col0 + m];
        v8f c0, c1, c2, c3;
#pragma unroll
        for (int v = 0; v < 8; ++v) { c0[v] = bv; c1[v] = bv; c2[v] = bv; c3[v] = bv; }
        gemm_tile4(ldsA, root, m, half, col0, c0, c1, c2, c3);
        if (full) {
            float* qp = Q + (size_t)(row0 + 8 * half) * 128 + col0 + m;
#pragma unroll
            for (int v = 0; v < 8; ++v) {
                qp[(size_t)(v     ) * 128] = c0[v];
                qp[(size_t)(v + 16) * 128] = c1[v];
                qp[(size_t)(v + 32) * 128] = c2[v];
                qp[(size_t)(v + 48) * 128] = c3[v];
            }
        } else {
#pragma unroll
            for (int v = 0; v < 8; ++v) {
                int r0 = row0 + v + 8 * half;
                if (r0      < N) Q[(size_t)(r0     ) * 128 + col0 + m] = c0[v];
                if (r0 + 16 < N) Q[(size_t)(r0 + 16) * 128 + col0 + m] = c1[v];
                if (r0 + 32 < N) Q[(size_t)(r0 + 32) * 128 + col0 + m] = c2[v];
                if (r0 + 48 < N) Q[(size_t)(r0 + 48) * 128 + col0 + m] = c3[v];
            }
        }
    }

    // ---- relation parts: XW[:, rel*128 + col] = P @ W_rel ----
    for (int rel = 0; rel < R; ++rel) {
        const float* B = Wrel + (size_t)rel * 128 * 128;
        v8f c0 = {}, c1 = {}, c2 = {}, c3 = {};
        gemm_tile4(ldsA, B, m, half, col0, c0, c1, c2, c3);
        int co = rel * 128 + col0 + m;
        if (full) {
            unsigned short* xp = XW + (size_t)(row0 + 8 * half) * ldx + co;
#pragma unroll
            for (int v = 0; v < 8; ++v) {
                xp[(size_t)(v     ) * ldx] = f2bf(c0[v]);
                xp[(size_t)(v + 16) * ldx] = f2bf(c1[v]);
                xp[(size_t)(v + 32) * ldx] = f2bf(c2[v]);
                xp[(size_t)(v + 48) * ldx] = f2bf(c3[v]);
            }
        } else {
#pragma unroll
            for (int v = 0; v < 8; ++v) {
                int r0 = row0 + v + 8 * half;
                if (r0      < N) XW[(size_t)(r0     ) * ldx + co] = f2bf(c0[v]);
                if (r0 + 16 < N) XW[(size_t)(r0 + 16) * ldx + co] = f2bf(c1[v]);
                if (r0 + 32 < N) XW[(size_t)(r0 + 32) * ldx + co] = f2bf(c2[v]);
                if (r0 + 48 < N) XW[(size_t)(r0 + 48) * ldx + co] = f2bf(c3[v]);
            }
        }
    }
}

// ---------------------------------------------------------------------------
// Aggregation + relu: one wave32 per dst node; lane holds 4 consecutive cols.
// Q[d] = relu(Q[d] + sum_e norm_e * XW[rel_e, src_e, :])
// ---------------------------------------------------------------------------
__global__ void agg_relu_kernel(const unsigned short* __restrict__ XW,
                                const int* __restrict__ offsets,
                                const int* __restrict__ ssrc,
                                const int* __restrict__ srel,
                                const float* __restrict__ snorm,
                                float* __restrict__ Q, int N, int R) {
    int gid  = blockIdx.x * blockDim.x + threadIdx.x;
    int node = gid >> 5;
    int lane = gid & 31;
    if (node >= N) return;
    int beg = offsets[node], end = offsets[node + 1];
    int ldx = R * 128;
    float a0 = 0.f, a1 = 0.f, a2 = 0.f, a3 = 0.f;
    for (int e = beg; e < end; ++e) {
        int   s  = ssrc[e];
        int   r  = srel[e];
        float nm = snorm[e];
        const uint2 d = *(const uint2*)(XW + (size_t)s * ldx + r * 128 + (lane << 2));
        a0 += nm * bf2f((unsigned short)(d.x & 0xFFFFu));
        a1 += nm * bf2f((unsigned short)(d.x >> 16));
        a2 += nm * bf2f((unsigned short)(d.y & 0xFFFFu));
        a3 += nm * bf2f((unsigned short)(d.y >> 16));
    }
    float4* qp = (float4*)(Q + (size_t)node * 128 + (lane << 2));
    float4 base = *qp;
    float4 o;
    o.x = fmaxf(base.x + a0, 0.f);
    o.y = fmaxf(base.y + a1, 0.f);
    o.z = fmaxf(base.z + a2, 0.f);
    o.w = fmaxf(base.w + a3, 0.f);
    *qp = o;
}

// ---------------------------------------------------------------------------
// Final FC: out[n] = h[n,:] . fc_w + fc_b ; wave per node, shuffle reduction
// ---------------------------------------------------------------------------
__global__ void fc_kernel(const float* __restrict__ Hh, const float* __restrict__ fw,
                          const float* __restrict__ fb, float* __restrict__ out, int N) {
    int gid  = blockIdx.x * blockDim.x + threadIdx.x;
    int node = gid >> 5;
    int lane = gid & 31;
    if (node >= N) return;
    float4 h = *(const float4*)(Hh + (size_t)node * 128 + (lane << 2));
    float4 w = *(const float4*)(fw + (lane << 2));
    float p = h.x * w.x + h.y * w.y + h.z * w.z + h.w * w.w;
    for (int off = 16; off > 0; off >>= 1) p += __shfl_down(p, off, 32);
    if (lane == 0) out[node] = p + fb[0];
}

// ---------------------------------------------------------------------------
extern "C" void kernel_launch(void* const* d_in, const int* in_sizes, int n_in,
                              void* d_out, int out_size, void* d_ws, size_t ws_size,
                              hipStream_t stream) {
    (void)n_in; (void)out_size; (void)ws_size;
    const float* x     = (const float*)d_in[0];
    const int*   ei    = (const int*)d_in[1];
    const int*   et    = (const int*)d_in[2];
    const float* W1    = (const float*)d_in[3];
    const float* root1 = (const float*)d_in[4];
    const float* b1    = (const float*)d_in[5];
    const float* W2    = (const float*)d_in[6];
    const float* root2 = (const float*)d_in[7];
    const float* b2    = (const float*)d_in[8];
    const float* W3    = (const float*)d_in[9];
    const float* root3 = (const float*)d_in[10];
    const float* b3    = (const float*)d_in[11];
    const float* fcw   = (const float*)d_in[12];
    const float* fcb   = (const float*)d_in[13];

    const int N = in_sizes[0] / 128;
    const int E = in_sizes[2];
    const int R = in_sizes[3] / (128 * 128);

    char* wsp = (char*)d_ws;
    auto alloc = [&](size_t bytes) -> char* {
        char* p = wsp;
        wsp += (bytes + 255) & ~(size_t)255;
        return p;
    };
    float*          deg     = (float*)alloc((size_t)N * R * 4);
    int*            cnt     = (int*)alloc((size_t)N * 4);
    int*            offsets = (int*)alloc((size_t)(N + 1) * 4);
    int*            cursor  = (int*)alloc((size_t)N * 4);
    int*            ssrc    = (int*)alloc((size_t)E * 4);
    int*            srel    = (int*)alloc((size_t)E * 4);
    float*          snorm   = (float*)alloc((size_t)E * 4);
    unsigned short* XW      = (unsigned short*)alloc((size_t)N * R * 128 * 2);
    float*          H1      = (float*)alloc((size_t)N * 128 * 4);
    float*          H2      = (float*)alloc((size_t)N * 128 * 4);

    hipMemsetAsync(deg, 0, (size_t)N * R * 4, stream);
    hipMemsetAsync(cnt, 0, (size_t)N * 4, stream);

    int eb = (E + 255) / 256;
    deg_cnt_kernel<<<eb, 256, 0, stream>>>(ei, et, deg, cnt, E, R);
    scan_kernel<<<1, 1024, 0, stream>>>(cnt, offsets, cursor, N);
    scatter_kernel<<<eb, 256, 0, stream>>>(ei, et, deg, cursor, ssrc, srel, snorm, E, R);

    int gblocks = (N + 63) / 64;
    int nwb     = (N * 32 + 255) / 256;   // wave-per-node kernels

    // layer 1: x -> H1
    gemm_layer_kernel<<<gblocks, 256, 0, stream>>>(x, W1, root1, b1, H1, XW, N, R);
    agg_relu_kernel<<<nwb, 256, 0, stream>>>(XW, offsets, ssrc, srel, snorm, H1, N, R);
    // layer 2: H1 -> H2
    gemm_layer_kernel<<<gblocks, 256, 0, stream>>>(H1, W2, root2, b2, H2, XW, N, R);
    agg_relu_kernel<<<nwb, 256, 0, stream>>>(XW, offsets, ssrc, srel, snorm, H2, N, R);
    // layer 3: H2 -> H1
    gemm_layer_kernel<<<gblocks, 256, 0, stream>>>(H2, W3, root3, b3, H1, XW, N, R);
    agg_relu_kernel<<<nwb, 256, 0, stream>>>(XW, offsets, ssrc, srel, snorm, H1, N, R);
    // final FC
    fc_kernel<<<nwb, 256, 0, stream>>>(H1, fcw, fcb, (float*)d_out, N);
}